// SelfAttention_9045201125483
// MI455X (gfx1250) — compile-verified
//
#include <hip/hip_runtime.h>

typedef __attribute__((ext_vector_type(2))) float v2f;
typedef __attribute__((ext_vector_type(8))) float v8f;
typedef __attribute__((ext_vector_type(4))) int   v4i;

#define B_   8
#define N_   1024
#define C_   768
#define H_   12
#define HD_  64
#define SCALE_ 0.036084391824351615f  /* 768^-0.5 */

static __device__ inline v8f wmma4(v2f a, v2f b, v8f c) {
  // V_WMMA_F32_16X16X4_F32 : D(16x16 f32) = A(16x4 f32) * B(4x16 f32) + C
  return __builtin_amdgcn_wmma_f32_16x16x4_f32(
      /*neg_a=*/false, a, /*neg_b=*/false, b,
      /*c_mod=*/(short)0, c, /*reuse_a=*/false, /*reuse_b=*/false);
}

// ---------------------------------------------------------------------------
// Async global -> LDS staging (CDNA5 GLOBAL_LOAD_ASYNC_TO_LDS_B128, ASYNCcnt).
// Falls back to a synchronous copy with identical pipeline structure if the
// builtins are not available in this toolchain.
// ---------------------------------------------------------------------------
#if defined(__has_builtin)
#if __has_builtin(__builtin_amdgcn_global_load_async_to_lds_b128) && \
    __has_builtin(__builtin_amdgcn_s_wait_asynccnt)
#define HAVE_ASYNC_LDS 1
#endif
#endif

typedef __attribute__((address_space(1))) v4i* gv4i_p;  // global v4i*
typedef __attribute__((address_space(3))) v4i* lv4i_p;  // LDS v4i*

static __device__ inline void stage_b128(const float* g, float* l) {
#ifdef HAVE_ASYNC_LDS
  __builtin_amdgcn_global_load_async_to_lds_b128((gv4i_p)(g), (lv4i_p)(l), 0, 0);
#else
  *(float4*)l = *(const float4*)g;
#endif
}

static __device__ inline void wait_stage() {
#ifdef HAVE_ASYNC_LDS
  __builtin_amdgcn_s_wait_asynccnt(0);
#endif
}

// ---------------------------------------------------------------------------
// Kernel 1: qkv = x @ W_qkv + b_qkv, scattered to Q/K/V in (B,H,N,HD) layout.
// Block = 256 threads (8 waves); block tile 64 (M) x 64 (N); wave tile 16x32.
// W tile (16K x 64N) staged in LDS via async b128, double-buffered.
// ---------------------------------------------------------------------------
__global__ __launch_bounds__(256) void qkv_gemm_kernel(
    const float* __restrict__ x, const float* __restrict__ W,
    const float* __restrict__ bias,
    float* __restrict__ qb, float* __restrict__ kb, float* __restrict__ vb) {
  __shared__ __align__(16) float Bs[2][16 * 64];

  const int tid  = threadIdx.x;
  const int wv   = tid >> 5;
  const int lane = tid & 31;
  const int lm   = lane & 15;     // tile row (A) / tile col (B,C)
  const int lk   = lane >> 4;     // half-wave selector for K slots
  const int rt   = wv >> 1;       // 0..3  -> M sub-tile
  const int cp   = wv & 1;        // 0..1  -> N sub-pair (32 cols)
  const int row0 = blockIdx.x * 64 + rt * 16;
  const int colB = blockIdx.y * 64;            // block column base
  const int colw = cp * 32;                    // wave column offset in tile

  // staging coordinates: thread t copies 4 floats of row (t>>4), cols (t&15)*4
  const int skr = tid >> 4;                    // 0..15 : k-row in chunk
  const int skc = (tid & 15) * 4;              // 0..60 : col in tile
  const float* Wg = W + (size_t)skr * (3 * C_) + colB + skc;
  float* Bl = &Bs[0][skr * 64 + skc];

  stage_b128(Wg, Bl);                          // chunk 0

  const float* Ap = x + (size_t)(row0 + lm) * C_ + lk * 2;
  v8f acc0 = {};
  v8f acc1 = {};
  int cur = 0;
  for (int c = 0; c < C_ / 16; ++c) {
    wait_stage();
    __syncthreads();                           // Bs[cur] ready for everyone
    if (c + 1 < C_ / 16)                       // prefetch next chunk
      stage_b128(Wg + (size_t)(c + 1) * 16 * (3 * C_), &Bs[cur ^ 1][skr * 64 + skc]);
    const float* bsc = Bs[cur];
#pragma unroll
    for (int kk = 0; kk < 4; ++kk) {
      v2f a = *(const v2f*)(Ap + c * 16 + kk * 4);
      const int kb = kk * 4 + lk * 2;
      v2f b0, b1;
      b0.x = bsc[kb * 64 + colw + lm];
      b0.y = bsc[(kb + 1) * 64 + colw + lm];
      b1.x = bsc[kb * 64 + colw + 16 + lm];
      b1.y = bsc[(kb + 1) * 64 + colw + 16 + lm];
      acc0 = wmma4(a, b0, acc0);
      acc1 = wmma4(a, b1, acc1);
    }
    cur ^= 1;
  }

#pragma unroll
  for (int st = 0; st < 2; ++st) {
    const v8f acc = st ? acc1 : acc0;
    const int colb = colB + colw + st * 16 + lm;  // column in [0, 2304)
    const int s    = colb / C_;                 // 0=q 1=k 2=v (uniform per tile)
    const int rem  = colb - s * C_;
    const int h    = rem >> 6;                  // head
    const int d    = rem & 63;                  // dim in head
    float* dst = (s == 0) ? qb : (s == 1) ? kb : vb;
    const float bv = bias[colb];
#pragma unroll
    for (int g = 0; g < 8; ++g) {
      const int row = row0 + g + lk * 8;        // global row = b*N + n
      const int b   = row >> 10;
      const int n   = row & (N_ - 1);
      dst[(((size_t)b * H_ + h) * N_ + n) * HD_ + d] = acc[g] + bv;
    }
  }
}

// ---------------------------------------------------------------------------
// Kernel 2: attention for one (b*h, 32-row q block).
// S tile (32 x 1024 f32) lives in LDS (128 KB of the 320 KB WGP LDS).
// ---------------------------------------------------------------------------
__global__ __launch_bounds__(256) void attn_kernel(
    const float* __restrict__ q, const float* __restrict__ k,
    const float* __restrict__ v, float* __restrict__ o) {
  __shared__ __align__(16) float P[32 * N_];   // exp(S - max), unnormalized
  __shared__ float pred[32 * 8];               // partial max / partial sum
  __shared__ float rowmax[32];
  __shared__ float rowinv[32];

  const int bh   = blockIdx.x;                 // 0..95  (b*H + h)
  const int qb   = blockIdx.y;                 // 0..31  (32-row q block)
  const int tid  = threadIdx.x;
  const int wv   = tid >> 5;
  const int lane = tid & 31;
  const int lm   = lane & 15;
  const int lk   = lane >> 4;

  // ---- stage 1: S = (Q K^T) * scale  -> LDS --------------------------------
  {
    const int rt = wv >> 2;                    // 0..1 : q row sub-tile
    const int cq = wv & 3;                     // 0..3 : 256-wide column strip
    const float* Qbase =
        q + ((size_t)bh * N_ + qb * 32 + rt * 16 + lm) * HD_ + lk * 2;
    v2f qa[16];                                // whole 16x64 Q tile as A-frags
#pragma unroll
    for (int ks = 0; ks < 16; ++ks) qa[ks] = *(const v2f*)(Qbase + ks * 4);

    for (int ct = 0; ct < 16; ++ct) {
      const int col = cq * 256 + ct * 16 + lm; // key index m
      const float* Kp = k + ((size_t)bh * N_ + col) * HD_ + lk * 2;
      v8f acc = {};
#pragma unroll
      for (int ks = 0; ks < 16; ++ks) {
        v2f bfr = *(const v2f*)(Kp + ks * 4);  // B[kb+j][m] = K[m][kb+j]
        acc = wmma4(qa[ks], bfr, acc);
      }
#pragma unroll
      for (int g = 0; g < 8; ++g) {
        const int r = rt * 16 + g + lk * 8;
        P[r * N_ + cq * 256 + ct * 16 + lm] = acc[g] * SCALE_;
      }
    }
  }
  __syncthreads();

  // ---- stage 2: row softmax (keep exp unnormalized, fold 1/sum into PV) ----
  {
    const int r    = tid >> 3;                 // 0..31
    const int seg  = tid & 7;                  // 8 threads per row, 128 cols
    const int base = r * N_ + seg * 128;
    float m = -__builtin_inff();
    for (int c2 = 0; c2 < 128; ++c2) m = fmaxf(m, P[base + c2]);
    pred[r * 8 + seg] = m;
    __syncthreads();
    if (tid < 32) {
      float mm = pred[tid * 8];
#pragma unroll
      for (int i = 1; i < 8; ++i) mm = fmaxf(mm, pred[tid * 8 + i]);
      rowmax[tid] = mm;
    }
    __syncthreads();
    const float rm = rowmax[r];
    float s = 0.f;
    for (int c2 = 0; c2 < 128; ++c2) {
      const float e = __expf(P[base + c2] - rm);
      P[base + c2] = e;
      s += e;
    }
    pred[r * 8 + seg] = s;
    __syncthreads();
    if (tid < 32) {
      float ss = 0.f;
#pragma unroll
      for (int i = 0; i < 8; ++i) ss += pred[tid * 8 + i];
      rowinv[tid] = 1.f / ss;
    }
    __syncthreads();
  }

  // ---- stage 3: O = P @ V, normalize, store to (B,N,C) ---------------------
  {
    const int rt = wv >> 2;                    // 0..1 : q row sub-tile
    const int dt = wv & 3;                     // 0..3 : 16-wide head-dim tile
    const float* Vbase = v + (size_t)bh * N_ * HD_ + dt * 16 + lm;
    const float* Prow  = &P[(rt * 16 + lm) * N_ + lk * 2];
    v8f acc = {};
#pragma unroll 4
    for (int m0 = 0; m0 < N_; m0 += 4) {
      v2f a = *(const v2f*)(Prow + m0);        // A[m_row=q][k=m]
      v2f bfr;
      bfr.x = Vbase[(size_t)(m0 + lk * 2) * HD_];
      bfr.y = Vbase[(size_t)(m0 + lk * 2 + 1) * HD_];
      acc = wmma4(a, bfr, acc);
    }
    const int bb = bh / H_;
    const int hh = bh - bb * H_;
#pragma unroll
    for (int g = 0; g < 8; ++g) {
      const int r = rt * 16 + g + lk * 8;
      const int n = qb * 32 + r;
      o[((size_t)bb * N_ + n) * C_ + hh * HD_ + dt * 16 + lm] =
          acc[g] * rowinv[r];
    }
  }
}

// ---------------------------------------------------------------------------
// Kernel 3: out = O @ W_proj + b_proj  (8192 x 768 x 768)
// Same async-LDS double-buffered B staging as kernel 1.
// ---------------------------------------------------------------------------
__global__ __launch_bounds__(256) void proj_gemm_kernel(
    const float* __restrict__ A, const float* __restrict__ W,
    const float* __restrict__ bias, float* __restrict__ out) {
  __shared__ __align__(16) float Bs[2][16 * 64];

  const int tid  = threadIdx.x;
  const int wv   = tid >> 5;
  const int lane = tid & 31;
  const int lm   = lane & 15;
  const int lk   = lane >> 4;
  const int rt   = wv >> 1;
  const int cp   = wv & 1;
  const int row0 = blockIdx.x * 64 + rt * 16;
  const int colB = blockIdx.y * 64;
  const int colw = cp * 32;

  const int skr = tid >> 4;
  const int skc = (tid & 15) * 4;
  const float* Wg = W + (size_t)skr * C_ + colB + skc;

  stage_b128(Wg, &Bs[0][skr * 64 + skc]);

  const float* Ap = A + (size_t)(row0 + lm) * C_ + lk * 2;
  v8f acc0 = {};
  v8f acc1 = {};
  int cur = 0;
  for (int c = 0; c < C_ / 16; ++c) {
    wait_stage();
    __syncthreads();
    if (c + 1 < C_ / 16)
      stage_b128(Wg + (size_t)(c + 1) * 16 * C_, &Bs[cur ^ 1][skr * 64 + skc]);
    const float* bsc = Bs[cur];
#pragma unroll
    for (int kk = 0; kk < 4; ++kk) {
      v2f a = *(const v2f*)(Ap + c * 16 + kk * 4);
      const int kb = kk * 4 + lk * 2;
      v2f b0, b1;
      b0.x = bsc[kb * 64 + colw + lm];
      b0.y = bsc[(kb + 1) * 64 + colw + lm];
      b1.x = bsc[kb * 64 + colw + 16 + lm];
      b1.y = bsc[(kb + 1) * 64 + colw + 16 + lm];
      acc0 = wmma4(a, b0, acc0);
      acc1 = wmma4(a, b1, acc1);
    }
    cur ^= 1;
  }

  const float bv0 = bias[colB + colw + lm];
  const float bv1 = bias[colB + colw + 16 + lm];
#pragma unroll
  for (int g = 0; g < 8; ++g) {
    const int row = row0 + g + lk * 8;
    out[(size_t)row * C_ + colB + colw + lm]      = acc0[g] + bv0;
    out[(size_t)row * C_ + colB + colw + 16 + lm] = acc1[g] + bv1;
  }
}

// ---------------------------------------------------------------------------
extern "C" void kernel_launch(void* const* d_in, const int* in_sizes, int n_in,
                              void* d_out, int out_size, void* d_ws,
                              size_t ws_size, hipStream_t stream) {
  (void)in_sizes; (void)n_in; (void)out_size; (void)ws_size;
  const float* x      = (const float*)d_in[0];
  const float* W_qkv  = (const float*)d_in[1];
  const float* b_qkv  = (const float*)d_in[2];
  const float* W_proj = (const float*)d_in[3];
  const float* b_proj = (const float*)d_in[4];
  float* out = (float*)d_out;

  float* ws = (float*)d_ws;
  const size_t per = (size_t)B_ * H_ * N_ * HD_;  // 6,291,456 floats (24 MB)
  float* qb = ws;
  float* kb = ws + per;
  float* vb = ws + 2 * per;
  float* ob = ws + 3 * per;

  dim3 blk(256);
  qkv_gemm_kernel<<<dim3((B_ * N_) / 64, (3 * C_) / 64), blk, 0, stream>>>(
      x, W_qkv, b_qkv, qb, kb, vb);
  attn_kernel<<<dim3(B_ * H_, N_ / 32), blk, 0, stream>>>(qb, kb, vb, ob);
  proj_gemm_kernel<<<dim3((B_ * N_) / 64, C_ / 64), blk, 0, stream>>>(
      ob, W_proj, b_proj, out);
}